// APA_22643067584516
// MI455X (gfx1250) — compile-verified
//
#include <hip/hip_runtime.h>
#include <hip/hip_bf16.h>

typedef __attribute__((ext_vector_type(2))) float v2f;
typedef __attribute__((ext_vector_type(8))) float v8f;

// ---------------------------------------------------------------- utilities
__global__ void fill_u32_kernel(unsigned int* p, unsigned int val, long n) {
    long i = (long)blockIdx.x * blockDim.x + threadIdx.x;
    if (i < n) p[i] = val;
}

// cnt[row[e]] += 1  (row-degree, as in the reference segment_sum over row)
__global__ void deg_kernel(const int* __restrict__ row, int* __restrict__ cnt, int E) {
    int e = blockIdx.x * blockDim.x + threadIdx.x;
    if (e < E) atomicAdd(&cnt[row[e]], 1);
}

// ---- 3-kernel exclusive scan over cnt[N] (N <= 65536) --------------------
__global__ void scan1_kernel(const int* __restrict__ cnt, int* __restrict__ off,
                             int* __restrict__ bsum, int N) {
    __shared__ int s[256];
    int i = blockIdx.x * 256 + threadIdx.x;
    int v = (i < N) ? cnt[i] : 0;
    s[threadIdx.x] = v;
    __syncthreads();
    for (int ofs = 1; ofs < 256; ofs <<= 1) {
        int t = (threadIdx.x >= ofs) ? s[threadIdx.x - ofs] : 0;
        __syncthreads();
        s[threadIdx.x] += t;
        __syncthreads();
    }
    if (i < N) off[i] = s[threadIdx.x] - v;          // exclusive within block
    if (threadIdx.x == 255) bsum[blockIdx.x] = s[255];
}

__global__ void scan2_kernel(int* __restrict__ bsum, int nb) {
    __shared__ int s[256];
    int v = (threadIdx.x < nb) ? bsum[threadIdx.x] : 0;
    s[threadIdx.x] = v;
    __syncthreads();
    for (int ofs = 1; ofs < 256; ofs <<= 1) {
        int t = (threadIdx.x >= ofs) ? s[threadIdx.x - ofs] : 0;
        __syncthreads();
        s[threadIdx.x] += t;
        __syncthreads();
    }
    if (threadIdx.x < nb) bsum[threadIdx.x] = s[threadIdx.x] - v;  // exclusive
}

__global__ void scan3_kernel(int* __restrict__ off, const int* __restrict__ bsum,
                             int* __restrict__ cursor, int N, int E) {
    int i = blockIdx.x * 256 + threadIdx.x;
    if (i < N) {
        int v = off[i] + bsum[blockIdx.x];
        off[i] = v;
        cursor[i] = v;
    }
    if (i == 0) off[N] = E;
}

// Scatter edges into row-sorted CSR order, packing (col, alpha*w) into int2.
// dinv(x) = x>0 ? rsqrt(x) : 0  (integer count => rsqrt(cnt)==rsqrt(max(cnt,1)))
__global__ void scatter_kernel(const int* __restrict__ row, const int* __restrict__ col,
                               const int* __restrict__ cnt, const float* __restrict__ alpha_p,
                               int* __restrict__ cursor, int2* __restrict__ scw, int E) {
    int e = blockIdx.x * blockDim.x + threadIdx.x;
    if (e >= E) return;
    int r = row[e], c = col[e];
    int pos = atomicAdd(&cursor[r], 1);
    float dr = (float)cnt[r];
    float dc = (float)cnt[c];
    float ir = (dr > 0.0f) ? rsqrtf(dr) : 0.0f;
    float ic = (dc > 0.0f) ? rsqrtf(dc) : 0.0f;
    int2 cw;
    cw.x = c;
    cw.y = __float_as_int(alpha_p[0] * ir * ic);
    scw[pos] = cw;
}

// kmap[known_idx[k]] = k (duplicates benign: oinit rows for duplicates identical)
__global__ void kmap_kernel(const int* __restrict__ kidx, int* __restrict__ kmap, int K) {
    int k = blockIdx.x * blockDim.x + threadIdx.x;
    if (k < K) kmap[kidx[k]] = k;
}

// out[known_idx[k], f] = x[known_idx[k], f];  oinit[k, f] = same
__global__ void init_known_kernel(const float* __restrict__ x, const int* __restrict__ kidx,
                                  float* __restrict__ out, float* __restrict__ oinit,
                                  long total, int d) {
    long i = (long)blockIdx.x * blockDim.x + threadIdx.x;
    if (i >= total) return;
    int k = (int)(i / d);
    int f = (int)(i % d);
    int n = kidx[k];
    float v = x[(long)n * d + f];
    out[(long)n * d + f] = v;
    oinit[i] = v;
}

// ------------------------------------------------- column sum via WMMA f32
// colsum[f] += sum_n src[n, f]  ==  ones(16x4) x B(4x16) accumulated.
// Column-separable result (A all-ones => D[i,j] = sum_k B[k,j]): out-of-range
// feature lanes load a clamped in-bounds address; their garbage stays in
// columns that are never flushed -> unconditional loads, no exec-mask churn.
__global__ void colsum_wmma_kernel(const float* __restrict__ src, float* __restrict__ colsum,
                                   int N, int d, int quads_per_chunk) {
    int ft    = blockIdx.x;          // feature tile (16 features)
    int chunk = blockIdx.y;          // row chunk
    int lane  = threadIdx.x;         // 0..31, one full wave per block
    int half  = lane >> 4;
    int j     = lane & 15;
    int f     = ft * 16 + j;
    bool fv   = (f < d);
    int fc    = fv ? f : (d - 1);    // clamped column for inactive lanes

    int Q  = N >> 2;
    int q0 = chunk * quads_per_chunk;
    int q1 = q0 + quads_per_chunk; if (q1 > Q) q1 = Q;

    v8f acc = {};
    v2f a; a.x = 1.0f; a.y = 1.0f;   // all-ones 16x4 A matrix

    const float* p     = src + (size_t)(q0 * 4 + half) * d + fc;
    const size_t step2 = (size_t)2 * d;
    const size_t step4 = (size_t)4 * d;

#pragma unroll 2
    for (int q = q0; q < q1; ++q) {
        v2f b;
        b.x = p[0];          // row r0+half
        b.y = p[step2];      // row r0+2+half
        p += step4;
        acc = __builtin_amdgcn_wmma_f32_16x16x4_f32(false, a, false, b,
                                                    (short)0, acc, false, false);
    }

    float tail = 0.0f;       // N % 4 leftovers (empty when N divisible by 4)
    if (chunk == gridDim.y - 1 && lane < 16 && fv) {
        for (int r = Q * 4; r < N; ++r) tail += src[(size_t)r * d + f];
    }

    // D layout: VGPR0, lanes 0..15 = row M=0 (all rows identical here).
    if (lane < 16 && fv) atomicAdd(&colsum[f], acc[0] + tail);
}

// Fused propagation + known-row reset, float4 over features:
//   acc = (1-alpha)/N * colsum + sum_e w_e * src[col_e]        (alpha in w_e)
//   if known: acc = beta*acc + (1-beta)*oinit[kmap[r]]
// One wave per row (blockDim.x=32, quads 0..24 active), b128 gathers/stores.
__global__ void prop_kernel(const int* __restrict__ off, const int2* __restrict__ scw,
                            const float* __restrict__ colsum,
                            const float* __restrict__ alpha_p, const float* __restrict__ beta_p,
                            const int* __restrict__ kmap, const float* __restrict__ oinit,
                            const float* __restrict__ src, float* __restrict__ dst,
                            int N, int d, float invN) {
    int q = threadIdx.x;                       // feature quad, active if 4*q < d
    int r = blockIdx.x * blockDim.y + threadIdx.y;
    int quads = (d + 3) >> 2;                  // 25 for d=100 (d divisible by 4)
    if (r >= N || q >= quads) return;

    float mixw = (1.0f - alpha_p[0]) * invN;
    const float4* cs4 = (const float4*)colsum;
    float4 m = cs4[q];
    float4 acc;
    acc.x = mixw * m.x; acc.y = mixw * m.y; acc.z = mixw * m.z; acc.w = mixw * m.w;

    int e0 = off[r];
    int e1 = off[r + 1];
    for (int e = e0; e < e1; ++e) {
        int2 cw = scw[e];                       // wave-uniform broadcast load
        float wgt = __int_as_float(cw.y);
        const float4* sp = (const float4*)(src + (size_t)cw.x * d);
        float4 v = sp[q];
        acc.x += wgt * v.x; acc.y += wgt * v.y;
        acc.z += wgt * v.z; acc.w += wgt * v.w;
    }

    int k = kmap[r];
    if (k >= 0) {
        float b  = beta_p[0];
        float ob = 1.0f - b;
        const float4* oi = (const float4*)(oinit + (size_t)k * d);
        float4 o = oi[q];
        acc.x = b * acc.x + ob * o.x; acc.y = b * acc.y + ob * o.y;
        acc.z = b * acc.z + ob * o.z; acc.w = b * acc.w + ob * o.w;
    }

    float4* dp = (float4*)(dst + (size_t)r * d);
    dp[q] = acc;
}

// ---------------------------------------------------------------- launcher
static inline size_t align_up(size_t v, size_t a) { return (v + a - 1) & ~(a - 1); }

extern "C" void kernel_launch(void* const* d_in, const int* in_sizes, int n_in,
                              void* d_out, int out_size, void* d_ws, size_t ws_size,
                              hipStream_t stream) {
    const int d = 100;
    const int N = in_sizes[0] / d;
    const int E = in_sizes[1] / 2;
    const int K = in_sizes[2];
    const int N_ITER = 30;

    const float* x       = (const float*)d_in[0];
    const int*   row     = (const int*)d_in[1];          // edge_index[0]
    const int*   col     = row + E;                      // edge_index[1]
    const int*   kidx    = (const int*)d_in[2];
    const float* alpha_p = (const float*)d_in[3];
    const float* beta_p  = (const float*)d_in[4];

    // workspace carving (~36 MB)
    char* w = (char*)d_ws;
    float* B      = (float*)w;  w += align_up((size_t)N * d * sizeof(float), 256);
    float* oinit  = (float*)w;  w += align_up((size_t)K * d * sizeof(float), 256);
    int2*  scw    = (int2*)w;   w += align_up((size_t)E * sizeof(int2), 256);
    int*   cnt    = (int*)w;    w += align_up((size_t)N * sizeof(int), 256);
    int*   off    = (int*)w;    w += align_up((size_t)(N + 1) * sizeof(int), 256);
    int*   cursor = (int*)w;    w += align_up((size_t)N * sizeof(int), 256);
    int*   kmap   = (int*)w;    w += align_up((size_t)N * sizeof(int), 256);
    int*   bsum   = (int*)w;    w += align_up(256 * sizeof(int), 256);
    float* colsum = (float*)w;  w += align_up(128 * sizeof(float), 256);

    float* A = (float*)d_out;   // ping-pong buffer A == final output location

    const long ND = (long)N * d;
    const long KD = (long)K * d;
    const int  scan_blocks = (N + 255) / 256;

    // --- one-time setup: degrees -> CSR build -> kmap -> init state --------
    fill_u32_kernel<<<(N + 255) / 256, 256, 0, stream>>>((unsigned int*)cnt, 0u, N);
    deg_kernel<<<(E + 255) / 256, 256, 0, stream>>>(row, cnt, E);
    scan1_kernel<<<scan_blocks, 256, 0, stream>>>(cnt, off, bsum, N);
    scan2_kernel<<<1, 256, 0, stream>>>(bsum, scan_blocks);
    scan3_kernel<<<scan_blocks, 256, 0, stream>>>(off, bsum, cursor, N, E);
    scatter_kernel<<<(E + 255) / 256, 256, 0, stream>>>(row, col, cnt, alpha_p,
                                                        cursor, scw, E);
    fill_u32_kernel<<<(N + 255) / 256, 256, 0, stream>>>((unsigned int*)kmap, 0xFFFFFFFFu, N);
    kmap_kernel<<<(K + 255) / 256, 256, 0, stream>>>(kidx, kmap, K);
    fill_u32_kernel<<<(int)((ND + 255) / 256), 256, 0, stream>>>((unsigned int*)A, 0u, ND);
    init_known_kernel<<<(int)((KD + 255) / 256), 256, 0, stream>>>(x, kidx, A, oinit, KD, d);

    // --- iterations: odd iters read A write B, even read B write A ----------
    // After 30 (even) iterations the result lands in A == d_out.
    const int ftiles  = (d + 15) / 16;                 // 7
    const int rchunks = 256;
    const int Q   = N >> 2;
    const int qpc = (Q + rchunks - 1) / rchunks;

    for (int it = 1; it <= N_ITER; ++it) {
        const float* src = (it & 1) ? A : B;
        float*       dst = (it & 1) ? B : A;

        fill_u32_kernel<<<1, 128, 0, stream>>>((unsigned int*)colsum, 0u, 128);
        colsum_wmma_kernel<<<dim3(ftiles, rchunks), 32, 0, stream>>>(src, colsum, N, d, qpc);
        prop_kernel<<<(N + 7) / 8, dim3(32, 8), 0, stream>>>(off, scw, colsum,
                                                             alpha_p, beta_p, kmap, oinit,
                                                             src, dst, N, d,
                                                             1.0f / (float)N);
    }
    (void)n_in; (void)out_size; (void)ws_size;
}